// DeltaFlowSeg_68977174774300
// MI455X (gfx1250) — compile-verified
//
#include <hip/hip_runtime.h>
#include <hip/hip_bf16.h>

// Problem constants (match reference)
#define BB 4
#define CVC 16
#define CPC 16
#define CCC 32
#define XD 256
#define YD 256
#define ZD 32
#define NVP 100000
#define NORIG 131072
#define NSLOT 50
#define DDIM 256
#define YZ (YD*ZD)             // 8192
#define XYZ (XD*YD*ZD)         // 2097152
#define BN (BB*NORIG)          // 524288 (column count of output)
#define NSLOT_PAD 64
#define TILES_PER_B (NVP/16)   // 6250

typedef float v2f __attribute__((ext_vector_type(2)));
typedef float v8f __attribute__((ext_vector_type(8)));

// ---------------- Kernel A1: h[b,o,n] = W1[o,:] . slot[b,n,:] + b1[o] ----------------
__global__ __launch_bounds__(256) void slot_fc1(const float* __restrict__ slot,
                                                const float* __restrict__ W1,
                                                const float* __restrict__ b1,
                                                float* __restrict__ h)
{
    int t = blockIdx.x * 256 + threadIdx.x;
    if (t >= BB * DDIM * NSLOT) return;
    int n = t % NSLOT;
    int o = (t / NSLOT) % DDIM;
    int b = t / (NSLOT * DDIM);
    const float* w = W1 + (size_t)o * DDIM;
    const float* s = slot + ((size_t)b * NSLOT + n) * DDIM;
    float acc = b1[o];
    for (int c = 0; c < DDIM; ++c) acc = fmaf(w[c], s[c], acc);
    h[t] = acc;   // layout [b][o][n]
}

// ---------------- Kernel A2: GroupNorm(4 groups over D) + ReLU, in place ----------------
__global__ __launch_bounds__(256) void slot_gnrelu(float* __restrict__ h,
                                                   const float* __restrict__ gamma,
                                                   const float* __restrict__ beta)
{
    const int b = blockIdx.x >> 2, g = blockIdx.x & 3;
    float* hp = h + ((size_t)b * DDIM + g * 64) * NSLOT;   // 64 channels x 50 slots, contiguous
    const int NEL = 64 * NSLOT;
    float s = 0.f, ss = 0.f;
    for (int i = threadIdx.x; i < NEL; i += 256) { float v = hp[i]; s += v; ss = fmaf(v, v, ss); }
    __shared__ float red[16];
    #pragma unroll
    for (int d = 1; d < 32; d <<= 1) { s += __shfl_xor(s, d, 32); ss += __shfl_xor(ss, d, 32); }
    const int w = threadIdx.x >> 5;
    if ((threadIdx.x & 31) == 0) { red[w] = s; red[8 + w] = ss; }
    __syncthreads();
    if (threadIdx.x == 0) {
        float S = 0.f, SS = 0.f;
        for (int i = 0; i < 8; ++i) { S += red[i]; SS += red[8 + i]; }
        red[0] = S; red[8] = SS;
    }
    __syncthreads();
    const float mu   = red[0] / (float)NEL;
    const float var  = red[8] / (float)NEL - mu * mu;
    const float rstd = rsqrtf(var + 1e-5f);
    for (int i = threadIdx.x; i < NEL; i += 256) {
        const int ch = g * 64 + i / NSLOT;
        float v = (hp[i] - mu) * rstd * gamma[ch] + beta[ch];
        hp[i] = fmaxf(v, 0.f);
    }
}

// -------- Kernel A3: slot_proj = W2 @ h + b2; L2-normalize per column; zero-pad to 64 --------
__global__ __launch_bounds__(32) void slot_proj_norm(const float* __restrict__ h,
                                                     const float* __restrict__ W2,
                                                     const float* __restrict__ b2,
                                                     float* __restrict__ snorm) // [B][32][64]
{
    const int n = blockIdx.x & 63;
    const int b = blockIdx.x >> 6;
    const int c = threadIdx.x;               // 0..31 output channel
    float sp = 0.f;
    if (n < NSLOT) {
        const float* w  = W2 + (size_t)c * DDIM;
        const float* hb = h + (size_t)b * DDIM * NSLOT + n;
        float acc = b2[c];
        for (int o = 0; o < DDIM; ++o) acc = fmaf(w[o], hb[(size_t)o * NSLOT], acc);
        sp = acc;
    }
    float ss = sp * sp;
    #pragma unroll
    for (int d = 1; d < 32; d <<= 1) ss += __shfl_xor(ss, d, 32);
    const float inv = 1.f / fmaxf(sqrtf(ss), 1e-12f);
    snorm[((size_t)b * CCC + c) * NSLOT_PAD + n] = (n < NSLOT) ? sp * inv : 0.f;
}

// ---------------- Kernel B: fill output with uniform softmax 1/50 ----------------
__global__ __launch_bounds__(256) void init_out(float* __restrict__ out)
{
    const size_t i  = (size_t)blockIdx.x * 256 + threadIdx.x;
    const size_t n4 = (size_t)NSLOT * BN / 4;
    if (i < n4) ((float4*)out)[i] = make_float4(0.02f, 0.02f, 0.02f, 0.02f);
}

// -------- Kernel C: gather feats, normalize, WMMA logits, softmax, scatter --------
__global__ __launch_bounds__(256) void mask_kernel(const float* __restrict__ vox,
                                                   const float* __restrict__ raw,
                                                   const int*   __restrict__ coords,
                                                   const int*   __restrict__ pidx,
                                                   const float* __restrict__ snorm,
                                                   float* __restrict__ out)
{
    __shared__ float s_sn[BB * CCC * NSLOT_PAD];   // 8192 floats = 32 KB
    for (int i = threadIdx.x; i < BB * CCC * NSLOT_PAD; i += 256) s_sn[i] = snorm[i];
    __syncthreads();

    const int wid  = (blockIdx.x * 256 + threadIdx.x) >> 5;  // exact grid: BB*TILES_PER_B waves
    const int lane = threadIdx.x & 31;
    const int b    = wid / TILES_PER_B;
    const int p0   = (wid % TILES_PER_B) * 16;
    const int m    = lane & 15;
    const int hh   = lane >> 4;        // half-wave id
    const int kh   = hh << 1;          // K sub-offset 0 or 2

    // ----- per-lane A fragment: features of point p0+m at K = {4s+kh, 4s+kh+1} -----
    const int  P   = p0 + m;
    const int* c3  = coords + ((size_t)b * NVP + P) * 3;
    const int  c0 = c3[0], c1 = c3[1], c2 = c3[2];
    const size_t lin = (size_t)c2 * YZ + (size_t)c1 * ZD + (size_t)c0;
    const float* voxb = vox + (size_t)b * CVC * XYZ + lin;
    const float* rawb = raw + ((size_t)b * NVP + P) * CPC;

    float av[16];
    float ss = 0.f;
    #pragma unroll
    for (int s = 0; s < 8; ++s) {
        const int k0 = 4 * s + kh;
        float x0, x1;
        if (s < 4) { x0 = voxb[(size_t)k0 * XYZ]; x1 = voxb[(size_t)(k0 + 1) * XYZ]; }
        else       { x0 = rawb[k0 - 16];          x1 = rawb[k0 - 15]; }
        av[2 * s] = x0; av[2 * s + 1] = x1;
        ss = fmaf(x0, x0, fmaf(x1, x1, ss));
    }
    ss += __shfl_xor(ss, 16, 32);                       // partner lane has other 16 feats
    const float scl = 20.0f / fmaxf(sqrtf(ss), 1e-12f); // L2-norm + 1/0.05 temperature

    v2f a[8];
    #pragma unroll
    for (int s = 0; s < 8; ++s) { a[s].x = av[2 * s] * scl; a[s].y = av[2 * s + 1] * scl; }

    // ----- logits tile [16 pts x 64 slots] via 8 K-steps x 4 N-tiles of f32 WMMA -----
    const float* sb = s_sn + (size_t)b * CCC * NSLOT_PAD;
    v8f acc[4] = {};
    #pragma unroll
    for (int s = 0; s < 8; ++s) {
        const int k0 = 4 * s + kh;
        #pragma unroll
        for (int j = 0; j < 4; ++j) {
            const int n = 16 * j + m;
            v2f bv;
            bv.x = sb[k0 * NSLOT_PAD + n];
            bv.y = sb[(k0 + 1) * NSLOT_PAD + n];
            acc[j] = __builtin_amdgcn_wmma_f32_16x16x4_f32(
                false, a[s], false, bv, (short)0, acc[j], false, false);
        }
    }

    // ----- softmax over 50 valid slots per point, scatter to out[slot, b*NORIG+idx] -----
    // C layout: acc[j][r] = logit(point p0+8*hh+r, slot 16*j+m); per-point values live
    // in one 16-lane half, so reduce with xor masks 1,2,4,8.
    const bool v3 = (m < 2);     // slot 48+m valid only when < 50
    #pragma unroll
    for (int r = 0; r < 8; ++r) {
        const int Pt = p0 + 8 * hh + r;
        const float x0 = acc[0][r], x1 = acc[1][r], x2 = acc[2][r], x3 = acc[3][r];
        float mx = fmaxf(fmaxf(x0, x1), x2);
        if (v3) mx = fmaxf(mx, x3);
        #pragma unroll
        for (int d = 1; d < 16; d <<= 1) mx = fmaxf(mx, __shfl_xor(mx, d, 32));
        const float e0 = expf(x0 - mx), e1 = expf(x1 - mx), e2 = expf(x2 - mx);
        const float e3 = v3 ? expf(x3 - mx) : 0.f;
        float sm = e0 + e1 + e2 + e3;
        #pragma unroll
        for (int d = 1; d < 16; d <<= 1) sm += __shfl_xor(sm, d, 32);
        const float rinv = 1.f / sm;
        const int idx = pidx[(size_t)b * NVP + Pt];
        float* ob = out + (size_t)b * NORIG + idx;
        ob[(size_t)(m)      * BN] = e0 * rinv;
        ob[(size_t)(m + 16) * BN] = e1 * rinv;
        ob[(size_t)(m + 32) * BN] = e2 * rinv;
        if (v3) ob[(size_t)(m + 48) * BN] = e3 * rinv;
    }
}

extern "C" void kernel_launch(void* const* d_in, const int* in_sizes, int n_in,
                              void* d_out, int out_size, void* d_ws, size_t ws_size,
                              hipStream_t stream)
{
    const float* vox    = (const float*)d_in[0];   // [B,CV,X,Y,Z]
    const float* raw    = (const float*)d_in[1];   // [B,NV,CP]
    const float* slot   = (const float*)d_in[2];   // [B,NSLOT,D]
    const float* W1     = (const float*)d_in[3];   // [D,D]
    const float* b1     = (const float*)d_in[4];   // [D]
    const float* gamma  = (const float*)d_in[5];   // [D]
    const float* beta   = (const float*)d_in[6];   // [D]
    const float* W2     = (const float*)d_in[7];   // [CC,D]
    const float* b2     = (const float*)d_in[8];   // [CC]
    const int*   coords = (const int*)d_in[9];     // [B,NV,3]
    const int*   pidx   = (const int*)d_in[10];    // [B,NV]
    float* out = (float*)d_out;                    // [NSLOT, B*NORIG]

    float* ws_h     = (float*)d_ws;                        // B*D*NSLOT floats
    float* ws_snorm = ws_h + (size_t)BB * DDIM * NSLOT;    // B*32*64 floats

    slot_fc1      <<<(BB * DDIM * NSLOT + 255) / 256, 256, 0, stream>>>(slot, W1, b1, ws_h);
    slot_gnrelu   <<<BB * 4, 256, 0, stream>>>(ws_h, gamma, beta);
    slot_proj_norm<<<BB * 64, 32, 0, stream>>>(ws_h, W2, b2, ws_snorm);
    init_out      <<<(int)(((size_t)NSLOT * BN / 4 + 255) / 256), 256, 0, stream>>>(out);
    mask_kernel   <<<(BB * TILES_PER_B) / 8, 256, 0, stream>>>(vox, raw, coords, pidx, ws_snorm, out);
}